// GATModel_14766097564207
// MI455X (gfx1250) — compile-verified
//
#include <hip/hip_runtime.h>
#include <math.h>

typedef float v2f __attribute__((ext_vector_type(2)));
typedef float v8f __attribute__((ext_vector_type(8)));

#define NEG_SLOPE 0.2f

// Order-preserving float <-> uint mapping for atomicMax-based segment max.
__device__ __forceinline__ unsigned fkey(float f) {
  unsigned u = __float_as_uint(f);
  return (u & 0x80000000u) ? ~u : (u | 0x80000000u);
}
__device__ __forceinline__ float fval(unsigned k) {
  unsigned u = (k & 0x80000000u) ? (k & 0x7FFFFFFFu) : ~k;
  return __uint_as_float(u);
}

__global__ void zero_kernel(unsigned* __restrict__ p, size_t n) {
  size_t i = (size_t)blockIdx.x * blockDim.x + threadIdx.x;
  if (i < n) p[i] = 0u;
}

// -------------------------------------------------------------------------
// Layer-1 GEMM: Hout[N,192] = X[N,128] @ W[128,192] using V_WMMA_F32_16X16X4_F32
// Block = 192 threads = 6 waves; each wave owns a 16x32 output tile.
// A-frag (16x4 f32): lane lp (0..15), half = lane>>4: a[r] = X[m0+lp][k0+2*half+r]
// B-frag (4x16 f32): b[r] = W[k0+2*half+r][col]
// D (16x16 f32, 8 VGPRs): acc[r] -> row m0 + r + 8*half, col = lp
// -------------------------------------------------------------------------
__global__ __launch_bounds__(192) void gemm1_wmma(const float* __restrict__ X,
                                                  const float* __restrict__ W,
                                                  float* __restrict__ Hout, int N) {
  const int wave = threadIdx.x >> 5;   // 0..5
  const int lane = threadIdx.x & 31;
  const int half = lane >> 4;
  const int lp   = lane & 15;
  const int m0   = blockIdx.x << 4;
  const int col  = wave * 32 + lp;

  int arow = m0 + lp; if (arow >= N) arow = N - 1;   // clamp (no divergence)
  const float* xrow = X + (size_t)arow * 128;

  v8f acc0 = {};
  v8f acc1 = {};
#pragma unroll 4
  for (int k0 = 0; k0 < 128; k0 += 4) {
    const int ka = k0 + 2 * half;
    v2f a, b0, b1;
    a.x  = xrow[ka];
    a.y  = xrow[ka + 1];
    b0.x = W[(size_t)ka * 192 + col];
    b0.y = W[(size_t)(ka + 1) * 192 + col];
    b1.x = W[(size_t)ka * 192 + col + 16];
    b1.y = W[(size_t)(ka + 1) * 192 + col + 16];
    acc0 = __builtin_amdgcn_wmma_f32_16x16x4_f32(false, a, false, b0, (short)0, acc0, false, false);
    acc1 = __builtin_amdgcn_wmma_f32_16x16x4_f32(false, a, false, b1, (short)0, acc1, false, false);
  }
#pragma unroll
  for (int r = 0; r < 8; ++r) {
    const int orow = m0 + r + 8 * half;
    if (orow < N) {
      Hout[(size_t)orow * 192 + col]      = acc0[r];
      Hout[(size_t)orow * 192 + col + 16] = acc1[r];
    }
  }
}

// Per-(node,head) attention logits: al = <h[n,h,:], a[h,:]>
__global__ void attn_logits(const float* __restrict__ feat, const float* __restrict__ asrc,
                            const float* __restrict__ adst, float* __restrict__ als,
                            float* __restrict__ ald, int N, int H, int C) {
  long long t = (long long)blockIdx.x * blockDim.x + threadIdx.x;
  if (t >= (long long)N * H) return;
  const int h = (int)(t % H);
  const long long n = t / H;
  const float* f = feat + ((size_t)n * H + (size_t)h) * C;
  float s = 0.f, d = 0.f;
  for (int c = 0; c < C; ++c) {
    const float v = f[c];
    s += v * asrc[h * C + c];
    d += v * adst[h * C + c];
  }
  als[t] = s;
  ald[t] = d;
}

__device__ __forceinline__ void edge_sd(const int* __restrict__ src, const int* __restrict__ dst,
                                        long long e, int E, int* s, int* d) {
  if (e < (long long)E) { *s = src[e]; *d = dst[e]; }
  else { *s = *d = (int)(e - E); }   // self loop
}

// Pass A: segment max of leaky_relu(al_s[src]+al_d[dst]) over incoming edges.
__global__ void edge_max(const int* __restrict__ src, const int* __restrict__ dst,
                         const float* __restrict__ als, const float* __restrict__ ald,
                         unsigned* __restrict__ emk, int E, int N, int H) {
  long long t = (long long)blockIdx.x * blockDim.x + threadIdx.x;
  if (t >= (long long)(E + N) * H) return;
  const int h = (int)(t % H);
  int s, d; edge_sd(src, dst, t / H, E, &s, &d);
  float v = als[(size_t)s * H + h] + ald[(size_t)d * H + h];
  v = v > 0.f ? v : NEG_SLOPE * v;
  atomicMax(&emk[(size_t)d * H + h], fkey(v));
}

// Pass B: denom = segment_sum(exp(e - emax[dst]))
__global__ void edge_sum(const int* __restrict__ src, const int* __restrict__ dst,
                         const float* __restrict__ als, const float* __restrict__ ald,
                         const unsigned* __restrict__ emk, float* __restrict__ den,
                         int E, int N, int H) {
  long long t = (long long)blockIdx.x * blockDim.x + threadIdx.x;
  if (t >= (long long)(E + N) * H) return;
  const int h = (int)(t % H);
  int s, d; edge_sd(src, dst, t / H, E, &s, &d);
  float v = als[(size_t)s * H + h] + ald[(size_t)d * H + h];
  v = v > 0.f ? v : NEG_SLOPE * v;
  const float ee = expf(v - fval(emk[(size_t)d * H + h]));
  atomicAdd(&den[(size_t)d * H + h], ee);
}

// Pass C: agg[dst,h,:] += alpha * feat[src,h,:]
__global__ void edge_agg(const int* __restrict__ src, const int* __restrict__ dst,
                         const float* __restrict__ als, const float* __restrict__ ald,
                         const unsigned* __restrict__ emk, const float* __restrict__ den,
                         const float* __restrict__ feat, float* __restrict__ agg,
                         int E, int N, int H, int C) {
  long long t = (long long)blockIdx.x * blockDim.x + threadIdx.x;
  if (t >= (long long)(E + N) * H) return;
  const int h = (int)(t % H);
  int s, d; edge_sd(src, dst, t / H, E, &s, &d);
  float v = als[(size_t)s * H + h] + ald[(size_t)d * H + h];
  v = v > 0.f ? v : NEG_SLOPE * v;
  const float alpha = expf(v - fval(emk[(size_t)d * H + h])) / den[(size_t)d * H + h];
  const float* f = feat + ((size_t)s * H + h) * C;
  float* o = agg + ((size_t)d * H + h) * C;
  for (int c = 0; c < C; ++c) atomicAdd(&o[c], alpha * f[c]);
}

// In-place: agg = relu(agg + bias[j % F])
__global__ void relu_bias(float* __restrict__ a, const float* __restrict__ b, size_t n, int F) {
  size_t i = (size_t)blockIdx.x * blockDim.x + threadIdx.x;
  if (i >= n) return;
  float v = a[i] + b[i % F];
  a[i] = v > 0.f ? v : 0.f;
}

// Layer-2 transform (192 -> 2) + attention logits (H=1, C=2)
__global__ void gemm2_logits(const float* __restrict__ h1r, const float* __restrict__ W2,
                             const float* __restrict__ a2s, const float* __restrict__ a2d,
                             float* __restrict__ h2, float* __restrict__ al2s,
                             float* __restrict__ al2d, int N) {
  int n = blockIdx.x * blockDim.x + threadIdx.x;
  if (n >= N) return;
  const float* r = h1r + (size_t)n * 192;
  float z0 = 0.f, z1 = 0.f;
  for (int j = 0; j < 192; ++j) {
    const float v = r[j];
    z0 += v * W2[2 * j];
    z1 += v * W2[2 * j + 1];
  }
  h2[(size_t)n * 2]     = z0;
  h2[(size_t)n * 2 + 1] = z1;
  al2s[n] = z0 * a2s[0] + z1 * a2s[1];
  al2d[n] = z0 * a2d[0] + z1 * a2d[1];
}

// out = log_softmax(agg2 + b2) over the 2 classes
__global__ void final_lsm(const float* __restrict__ agg2, const float* __restrict__ b2,
                          float* __restrict__ out, int N) {
  int n = blockIdx.x * blockDim.x + threadIdx.x;
  if (n >= N) return;
  const float z0 = agg2[(size_t)n * 2] + b2[0];
  const float z1 = agg2[(size_t)n * 2 + 1] + b2[1];
  const float m  = fmaxf(z0, z1);
  const float l  = m + logf(expf(z0 - m) + expf(z1 - m));
  out[(size_t)n * 2]     = z0 - l;
  out[(size_t)n * 2 + 1] = z1 - l;
}

extern "C" void kernel_launch(void* const* d_in, const int* in_sizes, int n_in,
                              void* d_out, int out_size, void* d_ws, size_t ws_size,
                              hipStream_t stream) {
  const float* x   = (const float*)d_in[0];
  const int*   ei  = (const int*)d_in[1];
  const float* W1  = (const float*)d_in[2];
  const float* a1s = (const float*)d_in[3];
  const float* a1d = (const float*)d_in[4];
  const float* b1  = (const float*)d_in[5];
  const float* W2  = (const float*)d_in[6];
  const float* a2s = (const float*)d_in[7];
  const float* a2d = (const float*)d_in[8];
  const float* b2  = (const float*)d_in[9];

  const int N = in_sizes[0] / 128;   // 50000
  const int E = in_sizes[1] / 2;     // 800000
  const int* src = ei;
  const int* dst = ei + E;

  // Workspace layout (floats / uints)
  float* p = (float*)d_ws;
  float* h1   = p; p += (size_t)N * 192;
  float* als1 = p; p += (size_t)N * 6;
  float* ald1 = p; p += (size_t)N * 6;
  float* h2   = p; p += (size_t)N * 2;
  float* al2s = p; p += (size_t)N;
  float* al2d = p; p += (size_t)N;
  unsigned* zbase = (unsigned*)p;                       // zero-init region start
  unsigned* emk1 = (unsigned*)p; p += (size_t)N * 6;
  float*    den1 = p;            p += (size_t)N * 6;
  float*    agg1 = p;            p += (size_t)N * 192;  // later reused as relu(h)
  unsigned* emk2 = (unsigned*)p; p += (size_t)N;
  float*    den2 = p;            p += (size_t)N;
  float*    agg2 = p;            p += (size_t)N * 2;
  const size_t zwords = (size_t)N * (6 + 6 + 192 + 1 + 1 + 2);

  const int TB = 256;
  zero_kernel<<<(unsigned)((zwords + TB - 1) / TB), TB, 0, stream>>>(zbase, zwords);

  // Layer 1
  gemm1_wmma<<<(unsigned)((N + 15) / 16), 192, 0, stream>>>(x, W1, h1, N);

  {
    long long tot = (long long)N * 6;
    attn_logits<<<(unsigned)((tot + TB - 1) / TB), TB, 0, stream>>>(h1, a1s, a1d, als1, ald1, N, 6, 32);
  }
  {
    long long tot = (long long)(E + N) * 6;
    unsigned g = (unsigned)((tot + TB - 1) / TB);
    edge_max<<<g, TB, 0, stream>>>(src, dst, als1, ald1, emk1, E, N, 6);
    edge_sum<<<g, TB, 0, stream>>>(src, dst, als1, ald1, emk1, den1, E, N, 6);
    edge_agg<<<g, TB, 0, stream>>>(src, dst, als1, ald1, emk1, den1, h1, agg1, E, N, 6, 32);
  }
  {
    size_t tot = (size_t)N * 192;
    relu_bias<<<(unsigned)((tot + TB - 1) / TB), TB, 0, stream>>>(agg1, b1, tot, 192);
  }

  // Layer 2
  gemm2_logits<<<(unsigned)((N + TB - 1) / TB), TB, 0, stream>>>(agg1, W2, a2s, a2d, h2, al2s, al2d, N);
  {
    long long tot = (long long)(E + N);
    unsigned g = (unsigned)((tot + TB - 1) / TB);
    edge_max<<<g, TB, 0, stream>>>(src, dst, al2s, al2d, emk2, E, N, 1);
    edge_sum<<<g, TB, 0, stream>>>(src, dst, al2s, al2d, emk2, den2, E, N, 1);
    edge_agg<<<g, TB, 0, stream>>>(src, dst, al2s, al2d, emk2, den2, h2, agg2, E, N, 1, 2);
  }

  final_lsm<<<(unsigned)((N + TB - 1) / TB), TB, 0, stream>>>(agg2, b2, (float*)d_out, N);
}